// MessagePassing_27797028340254
// MI455X (gfx1250) — compile-verified
//
#include <hip/hip_runtime.h>
#include <hip/hip_bf16.h>

// ---------------------------------------------------------------------------
// Equivariant message passing on MI455X (gfx1250, wave32, WMMA f32 16x16x4).
// All GEMMs run on the matrix pipe via V_WMMA_F32_16X16X4_F32 (fp32-exact).
// ---------------------------------------------------------------------------

#define NN 20000      // nodes
#define NE 320000     // edges

typedef float v2f __attribute__((ext_vector_type(2)));
typedef float v8f __attribute__((ext_vector_type(8)));

static __device__ __forceinline__ v8f wmma4(v2f a, v2f b, v8f c) {
  // D = A(16x4) * B(4x16) + C(16x16), fp32
  return __builtin_amdgcn_wmma_f32_16x16x4_f32(false, a, false, b, (short)0, c,
                                               false, false);
}

#define C_S 0.3826834323650898f
#define C_X 0.9238795325112867f

// ======================= node kernel: lin1 + sc ============================
// y      = [xs@W0/sqrt(m0) | per-comp xv@W1/sqrt(m1)]   (ldy = m0+3*m1)
// outacc = C_S * attr * [xs@scW0/sqrt(m0) | xv@scW1/sqrt(m1)] (ldo = m0o+3*m1o)
__global__ __launch_bounds__(256) void node_lin_kernel(
    const float* __restrict__ x, int ldx, const float* __restrict__ attr,
    const float* __restrict__ l1W0, const float* __restrict__ l1W1,
    const float* __restrict__ scW0, const float* __restrict__ scW1,
    float* __restrict__ y, int ldy, float* __restrict__ outacc, int ldo,
    int m0, int m1, int m0o, int m1o) {
  const int tid = threadIdx.x, wave = tid >> 5, lane = tid & 31;
  const int hi = lane >> 4, ln = lane & 15;
  const float inv0 = rsqrtf((float)m0), inv1 = rsqrtf((float)m1);
  const int ntiles = NN / 16;
  for (int t = blockIdx.x * 8 + wave; t < ntiles; t += gridDim.x * 8) {
    const int n0 = t * 16;
    const float* xr = x + (size_t)(n0 + ln) * ldx;
    // ---- scalar: ys = xs @ l1W0 ----
    for (int j = 0; j < m0; j += 16) {
      v8f acc = {};
      for (int k = 0; k < m0; k += 4) {
        v2f a = *(const v2f*)(xr + k + 2 * hi);
        const float* bp = l1W0 + (size_t)(k + 2 * hi) * m0 + j + ln;
        v2f b; b.x = bp[0] * inv0; b.y = bp[m0] * inv0;
        acc = wmma4(a, b, acc);
      }
      for (int r = 0; r < 8; r++)
        y[(size_t)(n0 + r + 8 * hi) * ldy + j + ln] = acc[r];
    }
    // ---- scalar: sc_s = xs @ scW0 * C_S * attr ----
    for (int j = 0; j < m0o; j += 16) {
      v8f acc = {};
      for (int k = 0; k < m0; k += 4) {
        v2f a = *(const v2f*)(xr + k + 2 * hi);
        const float* bp = scW0 + (size_t)(k + 2 * hi) * m0o + j + ln;
        v2f b; b.x = bp[0] * (C_S * inv0); b.y = bp[m0o] * (C_S * inv0);
        acc = wmma4(a, b, acc);
      }
      for (int r = 0; r < 8; r++) {
        int n = n0 + r + 8 * hi;
        outacc[(size_t)n * ldo + j + ln] = acc[r] * attr[n];
      }
    }
    // ---- vector parts, component c stride 3 ----
    for (int c = 0; c < 3; c++) {
      const float* xv = xr + m0 + c;
      for (int j = 0; j < m1; j += 16) {
        v8f acc = {};
        for (int k = 0; k < m1; k += 4) {
          int kk = k + 2 * hi;
          v2f a; a.x = xv[3 * kk]; a.y = xv[3 * kk + 3];
          const float* bp = l1W1 + (size_t)kk * m1 + j + ln;
          v2f b; b.x = bp[0] * inv1; b.y = bp[m1] * inv1;
          acc = wmma4(a, b, acc);
        }
        for (int r = 0; r < 8; r++)
          y[(size_t)(n0 + r + 8 * hi) * ldy + m0 + 3 * (j + ln) + c] = acc[r];
      }
      for (int j = 0; j < m1o; j += 16) {
        v8f acc = {};
        for (int k = 0; k < m1; k += 4) {
          int kk = k + 2 * hi;
          v2f a; a.x = xv[3 * kk]; a.y = xv[3 * kk + 3];
          const float* bp = scW1 + (size_t)kk * m1o + j + ln;
          v2f b; b.x = bp[0] * (C_S * inv1); b.y = bp[m1o] * (C_S * inv1);
          acc = wmma4(a, b, acc);
        }
        for (int r = 0; r < 8; r++) {
          int n = n0 + r + 8 * hi;
          outacc[(size_t)n * ldo + m0o + 3 * (j + ln) + c] = acc[r] * attr[n];
        }
      }
    }
  }
}

// ======================= edge kernel: MLP + messages + scatter =============
// Per 16-edge tile (one wave): h = silu(esc@W1*0.25); w = h@W2*0.125
// (w4 region gets extra 1/sqrt(3)); tensor-product messages scattered with
// fp32 atomics into agg[dst] (layout: [mid_s mk][mid_v mk*3], lda=4*mk).
__global__ __launch_bounds__(256) void edge_kernel(
    const float* __restrict__ esc, const float* __restrict__ eattr,
    const int* __restrict__ esrc, const int* __restrict__ edst,
    const float* __restrict__ y, int ldy, float* __restrict__ agg,
    const float* __restrict__ fcW1, const float* __restrict__ fcW2,
    int m0, int m1) {
  __shared__ __align__(16) float W1T[64 * 20];       // [col][k], padded
  __shared__ __align__(16) float W2T[384 * 68];      // [col][k], padded
  __shared__ __align__(16) float Hs[8][16 * 68];     // per-wave h tile [row][k]
  __shared__ __align__(16) float Meta[8][16 * 8];    // per-edge src,dst,sh0..shv2
  const int w2c = 2 * (m0 + m1);
  const int thr4 = 2 * m0 + m1;
  const int mk = m0 + m1, lda = 4 * mk;
  const int tid = threadIdx.x, wave = tid >> 5, lane = tid & 31;
  const int hi = lane >> 4, ln = lane & 15;

  for (int i = tid; i < 16 * 64; i += 256) {
    int k = i & 15, col = i >> 4;
    W1T[col * 20 + k] = fcW1[k * 64 + col] * 0.25f;   // fold 1/sqrt(16)
  }
  for (int i = tid; i < 64 * w2c; i += 256) {
    int col = i % w2c, k = i / w2c;
    float s = (col >= thr4) ? (0.125f * 0.57735026918962576f) : 0.125f;
    W2T[col * 68 + k] = fcW2[(size_t)k * w2c + col] * s;  // fold 1/8, 1/sqrt3
  }
  __syncthreads();

  const int ntiles = NE / 16;
  for (int t = blockIdx.x * 8 + wave; t < ntiles; t += gridDim.x * 8) {
    const int e0 = t * 16;
    if (lane < 16) {
      int e = e0 + lane;
      float* mp = &Meta[wave][lane * 8];
      mp[0] = __int_as_float(esrc[e]);
      mp[1] = __int_as_float(edst[e]);
      const float* ea = eattr + (size_t)e * 4;
      mp[2] = ea[0]; mp[3] = ea[1]; mp[4] = ea[2]; mp[5] = ea[3];
    }
    // A fragments of the 16x16 esc tile
    const float* ep = esc + (size_t)(e0 + ln) * 16 + 2 * hi;
    v2f A0 = *(const v2f*)(ep), A1 = *(const v2f*)(ep + 4);
    v2f A2 = *(const v2f*)(ep + 8), A3 = *(const v2f*)(ep + 12);
    // GEMM1: h(16x64) = esc @ W1, SiLU, park in LDS (A-layout relayout)
    for (int j = 0; j < 4; j++) {
      v8f acc = {};
      const float* wp = &W1T[(j * 16 + ln) * 20 + 2 * hi];
      v2f b;
      b.x = wp[0];  b.y = wp[1];  acc = wmma4(A0, b, acc);
      b.x = wp[4];  b.y = wp[5];  acc = wmma4(A1, b, acc);
      b.x = wp[8];  b.y = wp[9];  acc = wmma4(A2, b, acc);
      b.x = wp[12]; b.y = wp[13]; acc = wmma4(A3, b, acc);
#pragma unroll
      for (int r = 0; r < 8; r++) {
        float h = acc[r];
        h = h / (1.0f + __expf(-h));                      // SiLU
        Hs[wave][(r + 8 * hi) * 68 + j * 16 + ln] = h;
      }
    }
    asm volatile("s_wait_dscnt 0" ::: "memory");          // LDS producer->consumer
    // edge meta into registers
    int sA[8], dA[8]; float sh0[8], sv0[8], sv1[8], sv2[8];
#pragma unroll
    for (int r = 0; r < 8; r++) {
      const float* mp = &Meta[wave][(r + 8 * hi) * 8];
      sA[r] = __float_as_int(mp[0]); dA[r] = __float_as_int(mp[1]);
      sh0[r] = mp[2]; sv0[r] = mp[3]; sv1[r] = mp[4]; sv2[r] = mp[5];
    }
    // GEMM2: w(16 x w2c) = h @ W2, consume block-by-block
    const float* hrow = &Hs[wave][ln * 68 + 2 * hi];
    const int nblk = w2c / 16;
    for (int j = 0; j < nblk; j++) {
      v8f acc = {};
      const float* wp = &W2T[(j * 16 + ln) * 68 + 2 * hi];
      for (int k = 0; k < 64; k += 4) {
        v2f a = *(const v2f*)(hrow + k);
        v2f b = *(const v2f*)(wp + k);
        acc = wmma4(a, b, acc);
      }
      const int col = j * 16 + ln;   // region uniform across the wave
#pragma unroll
      for (int r = 0; r < 8; r++) {
        float w = acc[r];
        const float* yp = y + (size_t)sA[r] * ldy;
        float* ag = agg + (size_t)dA[r] * lda;
        if (col < m0) {                       // w1: mid_s[u] = w*es*sh0
          atomicAdd(&ag[col], w * yp[col] * sh0[r]);
        } else if (col < 2 * m0) {            // w2: mid_v[u][c] = w*es*shv[c]
          int u = col - m0;
          float base = w * yp[u];
          atomicAdd(&ag[mk + 3 * u + 0], base * sv0[r]);
          atomicAdd(&ag[mk + 3 * u + 1], base * sv1[r]);
          atomicAdd(&ag[mk + 3 * u + 2], base * sv2[r]);
        } else if (col < thr4) {              // w3: mid_v[m0+u][c] = w*ev[c]*sh0
          int u = col - 2 * m0;
          float b2 = w * sh0[r];
          const float* ev = yp + m0 + 3 * u;
          atomicAdd(&ag[mk + 3 * (m0 + u) + 0], b2 * ev[0]);
          atomicAdd(&ag[mk + 3 * (m0 + u) + 1], b2 * ev[1]);
          atomicAdd(&ag[mk + 3 * (m0 + u) + 2], b2 * ev[2]);
        } else {                              // w4: mid_s[m0+u] = w*(ev.shv)
          int u = col - thr4;
          const float* ev = yp + m0 + 3 * u;
          float dp = ev[0] * sv0[r] + ev[1] * sv1[r] + ev[2] * sv2[r];
          atomicAdd(&ag[m0 + u], w * dp);
        }
      }
    }
  }
}

// ======================= node kernel: lin2 + skip ==========================
// outp = acc_in + C_X/4/sqrt(mk) * [agg_s@W0 | per-comp agg_v@W1]
__global__ __launch_bounds__(256) void node_out_kernel(
    const float* __restrict__ agg, const float* __restrict__ l2W0,
    const float* __restrict__ l2W1, const float* __restrict__ acc_in,
    float* __restrict__ outp, int ldo, int m0, int m1, int m0o, int m1o) {
  const int mk = m0 + m1, lda = 4 * mk;
  const float sc = C_X * 0.25f * rsqrtf((float)mk);  // fold 1/sqrt(16) & C_X
  const int tid = threadIdx.x, wave = tid >> 5, lane = tid & 31;
  const int hi = lane >> 4, ln = lane & 15;
  const int ntiles = NN / 16;
  for (int t = blockIdx.x * 8 + wave; t < ntiles; t += gridDim.x * 8) {
    const int n0 = t * 16;
    const float* ar = agg + (size_t)(n0 + ln) * lda;
    for (int j = 0; j < m0o; j += 16) {
      v8f acc = {};
      for (int k = 0; k < mk; k += 4) {
        v2f a = *(const v2f*)(ar + k + 2 * hi);
        const float* bp = l2W0 + (size_t)(k + 2 * hi) * m0o + j + ln;
        v2f b; b.x = bp[0] * sc; b.y = bp[m0o] * sc;
        acc = wmma4(a, b, acc);
      }
      for (int r = 0; r < 8; r++) {
        size_t idx = (size_t)(n0 + r + 8 * hi) * ldo + j + ln;
        outp[idx] = acc_in[idx] + acc[r];
      }
    }
    for (int c = 0; c < 3; c++) {
      const float* av = ar + mk + c;
      for (int j = 0; j < m1o; j += 16) {
        v8f acc = {};
        for (int k = 0; k < mk; k += 4) {
          int kk = k + 2 * hi;
          v2f a; a.x = av[3 * kk]; a.y = av[3 * kk + 3];
          const float* bp = l2W1 + (size_t)kk * m1o + j + ln;
          v2f b; b.x = bp[0] * sc; b.y = bp[m1o] * sc;
          acc = wmma4(a, b, acc);
        }
        for (int r = 0; r < 8; r++) {
          size_t idx = (size_t)(n0 + r + 8 * hi) * ldo + m0o + 3 * (j + ln) + c;
          outp[idx] = acc_in[idx] + acc[r];
        }
      }
    }
  }
}

// ======================= gate (layers 0..2, fixed 192/64 -> 128/64) ========
__global__ __launch_bounds__(256) void gate_kernel(
    const float* __restrict__ in, int ldo, float* __restrict__ xout) {
  size_t i = (size_t)blockIdx.x * 256 + threadIdx.x;
  const size_t total = (size_t)NN * 320;
  if (i >= total) return;
  int n = (int)(i / 320), col = (int)(i % 320);
  const float* ip = in + (size_t)n * ldo;
  float o;
  if (col < 128) {
    float s = ip[col];
    o = s / (1.0f + __expf(-s));                 // silu
  } else {
    int u = (col - 128) / 3, c = (col - 128) % 3;
    float g = ip[128 + u];
    g = 1.0f / (1.0f + __expf(-g));              // sigmoid
    o = ip[192 + 3 * u + c] * g;
  }
  xout[i] = o;
}

// ======================= host driver =======================================
extern "C" void kernel_launch(void* const* d_in, const int* in_sizes, int n_in,
                              void* d_out, int out_size, void* d_ws,
                              size_t ws_size, hipStream_t stream) {
  (void)in_sizes; (void)n_in; (void)out_size; (void)ws_size;
  const float* node_features = (const float*)d_in[0];  // NN x 160
  const float* node_attr     = (const float*)d_in[1];  // NN
  const int*   edge_src      = (const int*)d_in[2];    // NE
  const int*   edge_dst      = (const int*)d_in[3];    // NE
  const float* edge_attr     = (const float*)d_in[4];  // NE x 4
  const float* edge_scalars  = (const float*)d_in[5];  // NE x 16

  static const int M0[4]  = {64, 128, 128, 128};
  static const int M1[4]  = {32, 64, 64, 64};
  static const int M0O[4] = {192, 192, 192, 64};
  static const int M1O[4] = {64, 64, 64, 32};

  float* ws = (float*)d_ws;
  float* X   = ws;                         // NN*320
  float* Y   = X + (size_t)NN * 320;       // NN*320
  float* AGG = Y + (size_t)NN * 320;       // NN*768
  float* OUT = AGG + (size_t)NN * 768;     // NN*384

  const float* x = node_features;
  int ldx = 160;
  for (int l = 0; l < 4; l++) {
    const int m0 = M0[l], m1 = M1[l], m0o = M0O[l], m1o = M1O[l];
    const int ldy = m0 + 3 * m1, ldo = m0o + 3 * m1o;
    const int pi = 6 + l * 8;
    const float* scW0  = (const float*)d_in[pi + 0];
    const float* scW1  = (const float*)d_in[pi + 1];
    const float* l1W0  = (const float*)d_in[pi + 2];
    const float* l1W1  = (const float*)d_in[pi + 3];
    const float* fcW1  = (const float*)d_in[pi + 4];
    const float* fcW2  = (const float*)d_in[pi + 5];
    const float* l2W0  = (const float*)d_in[pi + 6];
    const float* l2W1  = (const float*)d_in[pi + 7];

    node_lin_kernel<<<160, 256, 0, stream>>>(x, ldx, node_attr, l1W0, l1W1,
                                             scW0, scW1, Y, ldy, OUT, ldo, m0,
                                             m1, m0o, m1o);
    hipMemsetAsync(AGG, 0, (size_t)NN * 4 * (m0 + m1) * sizeof(float), stream);
    edge_kernel<<<400, 256, 0, stream>>>(edge_scalars, edge_attr, edge_src,
                                         edge_dst, Y, ldy, AGG, fcW1, fcW2, m0,
                                         m1);
    float* dst = (l == 3) ? (float*)d_out : OUT;
    node_out_kernel<<<160, 256, 0, stream>>>(AGG, l2W0, l2W1, OUT, dst, ldo,
                                             m0, m1, m0o, m1o);
    if (l < 3) {
      size_t total = (size_t)NN * 320;
      gate_kernel<<<(int)((total + 255) / 256), 256, 0, stream>>>(OUT, ldo, X);
      x = X;
      ldx = 320;
    }
  }
}